// RecurrentLayer_4277787427476
// MI455X (gfx1250) — compile-verified
//
#include <hip/hip_runtime.h>

// Problem constants (from the reference)
#define T_STEPS 1024
#define B_SZ    64
#define D_SZ    128
#define H_SZ    512
#define ALPHA   0.1f
#define NWG_SCAN 16            // persistent workgroups in the scan phase

typedef __attribute__((ext_vector_type(2))) float v2f;
typedef __attribute__((ext_vector_type(8))) float v8f;
typedef __attribute__((ext_vector_type(4))) unsigned int u32x4;
typedef __attribute__((ext_vector_type(8))) int i32x8;
typedef __attribute__((ext_vector_type(4))) int i32x4;

// ---------------------------------------------------------------------------
// Phase 0: zero the grid-barrier counter (deterministic per launch)
// ---------------------------------------------------------------------------
__global__ void rnn_init_kernel(unsigned* __restrict__ cnt) {
    if (threadIdx.x == 0) *cnt = 0u;
}

// ---------------------------------------------------------------------------
// Phase 1: w_in_u[t,b,h] = sum_d input[t,b,d] * W_in[h,d] + b_in[h]
// GEMM: M = T*B = 65536, N = H = 512, K = D = 128, fp32 via
// V_WMMA_F32_16X16X4_F32.  One wave computes a 16 x 64 strip (4 N-tiles)
// sharing its A fragments.  Result written directly into d_out.
// ---------------------------------------------------------------------------
__global__ __launch_bounds__(256) void rnn_proj_kernel(
    const float* __restrict__ inp,   // [T*B, D]
    const float* __restrict__ W_in,  // [H, D]
    const float* __restrict__ b_in,  // [H]
    float* __restrict__ out)         // [T*B, H]
{
    const int wave = threadIdx.x >> 5;
    const int lane = threadIdx.x & 31;
    const int half = lane >> 4;
    const int r    = lane & 15;

    const int gw    = blockIdx.x * 8 + wave;  // 0 .. 32767
    const int mtile = gw >> 3;                // 0 .. 4095
    const int ntq   = gw & 7;                 // 0 .. 7  (N strip of 4 tiles)
    const int m0    = mtile * 16;

    const float* ap = inp + (size_t)(m0 + r) * D_SZ + 2 * half;
    const float* bp = W_in + (size_t)(ntq * 64 + r) * D_SZ + 2 * half;

    v8f acc[4];
    acc[0] = (v8f){}; acc[1] = (v8f){}; acc[2] = (v8f){}; acc[3] = (v8f){};

    #pragma unroll 4
    for (int k0 = 0; k0 < D_SZ; k0 += 4) {
        v2f a  = *(const v2f*)(ap + k0);
        v2f b0 = *(const v2f*)(bp + k0 + 0 * 16 * D_SZ);
        v2f b1 = *(const v2f*)(bp + k0 + 1 * 16 * D_SZ);
        v2f b2 = *(const v2f*)(bp + k0 + 2 * 16 * D_SZ);
        v2f b3 = *(const v2f*)(bp + k0 + 3 * 16 * D_SZ);
        acc[0] = __builtin_amdgcn_wmma_f32_16x16x4_f32(false, a, false, b0, (short)0, acc[0], false, false);
        acc[1] = __builtin_amdgcn_wmma_f32_16x16x4_f32(false, a, false, b1, (short)0, acc[1], false, false);
        acc[2] = __builtin_amdgcn_wmma_f32_16x16x4_f32(false, a, false, b2, (short)0, acc[2], false, false);
        acc[3] = __builtin_amdgcn_wmma_f32_16x16x4_f32(false, a, false, b3, (short)0, acc[3], false, false);
    }

    #pragma unroll
    for (int j = 0; j < 4; ++j) {
        const int   n0   = ntq * 64 + j * 16;
        const float bias = b_in[n0 + r];
        #pragma unroll
        for (int i = 0; i < 8; ++i) {
            const int row = m0 + i + 8 * half;
            out[(size_t)row * H_SZ + (n0 + r)] = acc[j][i] + bias;
        }
    }
}

// ---------------------------------------------------------------------------
// Phase 2: persistent scan.
//   v_t  = (1-a)*v_{t-1} + a*( fr_{t-1} @ W_hid^T + b_hid + w_in_u[t] )
//   fr_t = relu(v_t)   -> overwrites d_out[t] in place
//
// 16 WGs x 8 waves = 128 waves = 128 (16x16) tiles of [B=64, H=512].
// Each WG's 32 W_hid rows are DMA'd into LDS once by the Tensor Data Mover
// (with TDM LDS padding: +2 DWORDs every 256 -> row stride 516 floats,
// which makes the per-lane ds_load_b64 B-fragment reads bank-conflict-free).
// v stays in 8 accumulator VGPRs per lane for the whole kernel.
// Steps separated by an agent-scope atomic grid barrier.
// ---------------------------------------------------------------------------
#define WL_STRIDE 516          // 512 + 2 pad DWORDs after each 256 DWORDs

__global__ __launch_bounds__(256) void rnn_scan_kernel(
    const float* __restrict__ W_hid,  // [H, H]
    const float* __restrict__ b_hid,  // [H]
    float* __restrict__ out,          // [T, B, H]: in = w_in_u, out = fr
    unsigned* __restrict__ barrier_cnt)
{
    __shared__ float Wl[32 * WL_STRIDE];

    const int tid  = threadIdx.x;
    const int wave = tid >> 5;
    const int lane = tid & 31;
    const int half = lane >> 4;
    const int r    = lane & 15;

    const int blk   = blockIdx.x;          // 0..15
    const int gwave = blk * 8 + wave;      // 0..127
    const int mt    = gwave & 3;           // 0..3   (B tiles)
    const int nt    = gwave >> 2;          // 0..31  (H tiles); nt in {2*blk, 2*blk+1}
    const int m0    = mt * 16;
    const int n0    = nt * 16;
    const int nbase = blk * 32;            // W_hid rows cached by this WG
    const int nloc  = n0 - nbase;          // 0 or 16

#if __has_builtin(__builtin_amdgcn_tensor_load_to_lds)
    // ---- TDM: DMA W_hid rows [nbase, nbase+32) x 512 f32 into LDS ----
    if (wave == 0) {
        const unsigned long long gaddr =
            (unsigned long long)(uintptr_t)(W_hid + (size_t)nbase * H_SZ);
        const unsigned lds_base = (unsigned)(uintptr_t)(&Wl[0]); // low 32b = LDS offset

        // D# group 0: count=1 | lds_addr | global_addr(57b) | type=2
        u32x4 g0 = { 1u,
                     lds_base,
                     (unsigned)(gaddr & 0xFFFFFFFFu),
                     (unsigned)((gaddr >> 32) & 0x01FFFFFFu) | (2u << 30) };
        // D# group 1:
        //  w0: data_size=2(4B) | pad_enable | pad_interval=7(256dw) | pad_amount=1(2dw)
        //  w1: abar_addr=0 | tensor_dim0.lo=512
        //  w2: tensor_dim0.hi=0 | tensor_dim1.lo=512
        //  w3: tensor_dim1.hi=0 | tile_dim0=512
        //  w4: tile_dim1=32 | tile_dim2=0
        //  w5: tensor_dim0_stride=512 ; w6,w7: dim1 stride (unused for 2D) = 0
        i32x8 g1 = { (int)((2u << 16) | (1u << 20) | (7u << 22) | (1u << 25)),
                     (int)(512u << 16),
                     (int)(512u << 16),
                     (int)(512u << 16),
                     32, 512, 0, 0 };
        i32x4 g2 = { 0, 0, 0, 0 };
        i32x4 g3 = { 0, 0, 0, 0 };
#if __has_include(<hip/amd_detail/amd_gfx1250_TDM.h>)
        i32x8 g4 = { 0, 0, 0, 0, 0, 0, 0, 0 };
        __builtin_amdgcn_tensor_load_to_lds(g0, g1, g2, g3, g4, 0);
#else
        __builtin_amdgcn_tensor_load_to_lds(g0, g1, g2, g3, 0);
#endif
        __builtin_amdgcn_s_wait_tensorcnt(0);
    }
    __syncthreads();
#else
    // Fallback: manual preload into the same padded layout.
    for (int idx = tid; idx < 32 * H_SZ; idx += 256) {
        const int n = idx >> 9;
        const int k = idx & (H_SZ - 1);
        Wl[n * WL_STRIDE + k + 2 * (k >= 256)] =
            W_hid[(size_t)(nbase + n) * H_SZ + k];
    }
    __syncthreads();
#endif

    const float bh = b_hid[n0 + r];
    const float* bptr = &Wl[(nloc + r) * WL_STRIDE + 2 * half];

    v8f v = (v8f){};   // membrane state, persistent in VGPRs

    for (int t = 0; t < T_STEPS; ++t) {
        v8f c = (v8f){};
        if (t > 0) {
            // A = fr_{t-1} read from out[t-1]; B = W_hid slice from LDS.
            const float* aptr = out + (size_t)(t - 1) * (B_SZ * H_SZ)
                                    + (size_t)(m0 + r) * H_SZ + 2 * half;
            #pragma unroll 4
            for (int k0 = 0; k0 < 256; k0 += 4) {
                v2f a = *(const v2f*)(aptr + k0);
                v2f b = *(const v2f*)(bptr + k0);
                c = __builtin_amdgcn_wmma_f32_16x16x4_f32(false, a, false, b, (short)0, c, false, false);
            }
            #pragma unroll 4
            for (int k0 = 256; k0 < 512; k0 += 4) {
                v2f a = *(const v2f*)(aptr + k0);
                v2f b = *(const v2f*)(bptr + k0 + 2);   // skip mid-row pad DWORDs
                c = __builtin_amdgcn_wmma_f32_16x16x4_f32(false, a, false, b, (short)0, c, false, false);
            }
        } // t==0: fr_0 = relu(0) = 0 -> recurrent term is 0.

        float* outT = out + (size_t)t * (B_SZ * H_SZ);
        #pragma unroll
        for (int i = 0; i < 8; ++i) {
            const int row = m0 + i + 8 * half;
            const size_t off = (size_t)row * H_SZ + (n0 + r);
            const float u  = outT[off];                      // w_in_u[t] (has b_in)
            const float vn = (1.0f - ALPHA) * v[i] + ALPHA * (c[i] + bh + u);
            v[i] = vn;
            outT[off] = vn > 0.0f ? vn : 0.0f;               // fr_t
        }

        // ---- grid-wide barrier: make fr_t visible to all WGs ----
        __threadfence();           // release our stores to device scope
        __syncthreads();
        if (tid == 0) {
            __hip_atomic_fetch_add(barrier_cnt, 1u, __ATOMIC_RELEASE,
                                   __HIP_MEMORY_SCOPE_AGENT);
            const unsigned target = (unsigned)(t + 1) * NWG_SCAN;
            while (__hip_atomic_load(barrier_cnt, __ATOMIC_ACQUIRE,
                                     __HIP_MEMORY_SCOPE_AGENT) < target) {
                __builtin_amdgcn_s_sleep(8);
            }
        }
        __syncthreads();
        __threadfence();           // acquire: invalidate stale near-cache lines
    }
}

// ---------------------------------------------------------------------------
extern "C" void kernel_launch(void* const* d_in, const int* in_sizes, int n_in,
                              void* d_out, int out_size, void* d_ws, size_t ws_size,
                              hipStream_t stream) {
    const float* inp   = (const float*)d_in[0];  // [T,B,D]
    const float* W_in  = (const float*)d_in[1];  // [H,D]
    const float* b_in  = (const float*)d_in[2];  // [H]
    const float* W_hid = (const float*)d_in[3];  // [H,H]
    const float* b_hid = (const float*)d_in[4];  // [H]
    float* out = (float*)d_out;                  // [T,B,H]
    unsigned* cnt = (unsigned*)d_ws;             // 4 bytes of scratch

    rnn_init_kernel<<<1, 64, 0, stream>>>(cnt);
    // 32768 waves (4096 M-tiles x 8 N-strips), 8 waves per block
    rnn_proj_kernel<<<4096, 256, 0, stream>>>(inp, W_in, b_in, out);
    // persistent scan: 16 WGs x 256 threads
    rnn_scan_kernel<<<NWG_SCAN, 256, 0, stream>>>(W_hid, b_hid, out, cnt);
}